// PytorchLSTM_23467701305597
// MI455X (gfx1250) — compile-verified
//
#include <hip/hip_runtime.h>
#include <hip/hip_bf16.h>

typedef float v2f __attribute__((ext_vector_type(2)));
typedef float v8f __attribute__((ext_vector_type(8)));

namespace {
constexpr int kH   = 40;    // hidden dim
constexpr int kG   = 160;   // 4*H gates
constexpr int kT   = 256;   // sequence length
constexpr int kDin = 3;     // input feature dim
constexpr int kMT  = 16;    // batch rows per block (one WMMA M tile)
constexpr int kNTW = 5;     // N-tiles per wave (10 tiles of 16 cols / 2 waves)
constexpr int kKS  = 10;    // K steps of 4 covering H=40

__device__ __forceinline__ float sigm(float v) { return 1.0f / (1.0f + __expf(-v)); }
} // namespace

__global__ __launch_bounds__(64, 1)
void lstm2_fused_wmma(const float* __restrict__ x,
                      const float* __restrict__ W_ih1,
                      const float* __restrict__ W_hh1,
                      const float* __restrict__ b1,
                      const float* __restrict__ W_ih2,
                      const float* __restrict__ W_hh2,
                      const float* __restrict__ b2,
                      const float* __restrict__ W_fc,
                      const float* __restrict__ b_fc,
                      float* __restrict__ out)
{
    // Per-block recurrent state + gate staging, all in LDS (20.5 KB).
    __shared__ __align__(16) float sh_h1[kMT * kH];
    __shared__ __align__(16) float sh_c1[kMT * kH];
    __shared__ __align__(16) float sh_h2[kMT * kH];
    __shared__ __align__(16) float sh_c2[kMT * kH];
    __shared__ __align__(16) float sh_g [kMT * kG];
    __shared__ __align__(16) float sh_x [kMT * kDin];

    const int tid  = threadIdx.x;
    const int lane = tid & 31;
    const int wave = tid >> 5;      // 2 waves split the 10 N-tiles
    const int lh   = lane >> 4;     // half-wave select (WMMA lane layout)
    const int ll   = lane & 15;
    const int b0   = blockIdx.x * kMT;

    for (int e = tid; e < kMT * kH; e += 64) {
        sh_h1[e] = 0.f; sh_c1[e] = 0.f; sh_h2[e] = 0.f; sh_c2[e] = 0.f;
    }

    // Per-lane per-N-tile constants: bias and the tiny W_ih1 column (Din=3).
    float b1c[kNTW], b2c[kNTW], wi1[kNTW][kDin];
    #pragma unroll
    for (int i = 0; i < kNTW; ++i) {
        const int col = (wave * kNTW + i) * 16 + ll;   // gate column 0..159
        b1c[i] = b1[col];
        b2c[i] = b2[col];
        wi1[i][0] = W_ih1[col * kDin + 0];
        wi1[i][1] = W_ih1[col * kDin + 1];
        wi1[i][2] = W_ih1[col * kDin + 2];
    }
    __syncthreads();

    for (int t = 0; t < kT; ++t) {
        // Stage x[b0..b0+15][t][:] (48 floats) into LDS.
        if (tid < kMT * kDin) {
            const int m = tid / kDin, d = tid % kDin;
            sh_x[tid] = x[((size_t)(b0 + m) * kT + t) * kDin + d];
        }
        __syncthreads();

        // ---------------- layer 1: gates = b1 + x*W_ih1^T + h1*W_hh1^T ------
        v8f acc[kNTW];
        #pragma unroll
        for (int i = 0; i < kNTW; ++i) {
            #pragma unroll
            for (int r = 0; r < 8; ++r) {
                const int m = r + 8 * lh;              // C-tile row for this VGPR
                acc[i][r] = b1c[i]
                          + wi1[i][0] * sh_x[m * kDin + 0]
                          + wi1[i][1] * sh_x[m * kDin + 1]
                          + wi1[i][2] * sh_x[m * kDin + 2];
            }
        }
        #pragma unroll
        for (int ks = 0; ks < kKS; ++ks) {
            const int k = ks * 4 + 2 * lh;
            const v2f a = *(const v2f*)&sh_h1[ll * kH + k];          // A 16x4 frag
            #pragma unroll
            for (int i = 0; i < kNTW; ++i) {
                const int n = (wave * kNTW + i) * 16 + ll;
                const v2f bf = *(const v2f*)&W_hh1[n * kH + k];      // B 4x16 frag (W^T)
                acc[i] = __builtin_amdgcn_wmma_f32_16x16x4_f32(
                    false, a, false, bf, (short)0, acc[i], false, false);
            }
        }
        #pragma unroll
        for (int i = 0; i < kNTW; ++i) {
            const int col = (wave * kNTW + i) * 16 + ll;
            #pragma unroll
            for (int r = 0; r < 8; ++r)
                sh_g[(r + 8 * lh) * kG + col] = acc[i][r];
        }
        __syncthreads();

        // layer 1 elementwise: i,f,g,o -> c,h   (640 elems / 64 threads)
        #pragma unroll
        for (int j = 0; j < 10; ++j) {
            const int e  = tid + 64 * j;
            const int m  = e / kH, hj = e % kH;
            const float gi = sigm (sh_g[m * kG + 0 * kH + hj]);
            const float gf = sigm (sh_g[m * kG + 1 * kH + hj]);
            const float gg = tanhf(sh_g[m * kG + 2 * kH + hj]);
            const float go = sigm (sh_g[m * kG + 3 * kH + hj]);
            const float c  = gf * sh_c1[m * kH + hj] + gi * gg;
            sh_c1[m * kH + hj] = c;
            sh_h1[m * kH + hj] = go * tanhf(c);
        }
        __syncthreads();

        // ---------------- layer 2: gates = b2 + h1*W_ih2^T + h2*W_hh2^T -----
        v8f acc2[kNTW];
        #pragma unroll
        for (int i = 0; i < kNTW; ++i)
            #pragma unroll
            for (int r = 0; r < 8; ++r)
                acc2[i][r] = b2c[i];
        #pragma unroll
        for (int ks = 0; ks < kKS; ++ks) {
            const int k = ks * 4 + 2 * lh;
            const v2f a = *(const v2f*)&sh_h1[ll * kH + k];
            #pragma unroll
            for (int i = 0; i < kNTW; ++i) {
                const int n = (wave * kNTW + i) * 16 + ll;
                const v2f bf = *(const v2f*)&W_ih2[n * kH + k];
                acc2[i] = __builtin_amdgcn_wmma_f32_16x16x4_f32(
                    false, a, false, bf, (short)0, acc2[i], false, false);
            }
        }
        #pragma unroll
        for (int ks = 0; ks < kKS; ++ks) {
            const int k = ks * 4 + 2 * lh;
            const v2f a = *(const v2f*)&sh_h2[ll * kH + k];
            #pragma unroll
            for (int i = 0; i < kNTW; ++i) {
                const int n = (wave * kNTW + i) * 16 + ll;
                const v2f bf = *(const v2f*)&W_hh2[n * kH + k];
                acc2[i] = __builtin_amdgcn_wmma_f32_16x16x4_f32(
                    false, a, false, bf, (short)0, acc2[i], false, false);
            }
        }
        #pragma unroll
        for (int i = 0; i < kNTW; ++i) {
            const int col = (wave * kNTW + i) * 16 + ll;
            #pragma unroll
            for (int r = 0; r < 8; ++r)
                sh_g[(r + 8 * lh) * kG + col] = acc2[i][r];
        }
        __syncthreads();

        // layer 2 elementwise
        #pragma unroll
        for (int j = 0; j < 10; ++j) {
            const int e  = tid + 64 * j;
            const int m  = e / kH, hj = e % kH;
            const float gi = sigm (sh_g[m * kG + 0 * kH + hj]);
            const float gf = sigm (sh_g[m * kG + 1 * kH + hj]);
            const float gg = tanhf(sh_g[m * kG + 2 * kH + hj]);
            const float go = sigm (sh_g[m * kG + 3 * kH + hj]);
            const float c  = gf * sh_c2[m * kH + hj] + gi * gg;
            sh_c2[m * kH + hj] = c;
            sh_h2[m * kH + hj] = go * tanhf(c);
        }
        __syncthreads();
    }

    // Final FC: out[b][o] = b_fc[o] + sum_j h2[b][j] * W_fc[o][j]
    if (tid < 2 * kMT) {
        const int m = tid & 15, o = tid >> 4;
        float s = b_fc[o];
        #pragma unroll
        for (int j = 0; j < kH; ++j)
            s += sh_h2[m * kH + j] * W_fc[o * kH + j];
        out[(size_t)(b0 + m) * 2 + o] = s;
    }
}

extern "C" void kernel_launch(void* const* d_in, const int* in_sizes, int n_in,
                              void* d_out, int out_size, void* d_ws, size_t ws_size,
                              hipStream_t stream) {
    const float* x     = (const float*)d_in[0];
    const float* W_ih1 = (const float*)d_in[1];
    const float* W_hh1 = (const float*)d_in[2];
    const float* b1    = (const float*)d_in[3];
    const float* W_ih2 = (const float*)d_in[4];
    const float* W_hh2 = (const float*)d_in[5];
    const float* b2    = (const float*)d_in[6];
    const float* W_fc  = (const float*)d_in[7];
    const float* b_fc  = (const float*)d_in[8];
    float* out = (float*)d_out;

    const int B = in_sizes[0] / (kT * kDin);   // 8192
    const int grid = B / kMT;                  // 512 blocks, 2 waves each

    lstm2_fused_wmma<<<grid, 64, 0, stream>>>(
        x, W_ih1, W_hh1, b1, W_ih2, W_hh2, b2, W_fc, b_fc, out);
}